// VAE_Baseline_21680994910278
// MI455X (gfx1250) — compile-verified
//
#include <hip/hip_runtime.h>
#include <math.h>
#include <stdint.h>

// Problem shape (fixed by reference setup_inputs)
#define S_ 3
#define B_ 256
#define T_ 512
#define D_ 64
#define TD_ (T_ * D_)

// Tiling: 16 t-rows x 64 d = 1024 floats (4 KB) per buffer per tile.
#define NTILE 32                 // T_/16
#define TILE_FLOATS 1024
#define STAGE_FLOATS (5 * TILE_FLOATS)   // truth, mask, pred0..2
#define NSTAGE 4                 // quad buffer -> 80 KB LDS, 3 tiles in flight

typedef __attribute__((ext_vector_type(2))) float v2f;
typedef __attribute__((ext_vector_type(8))) float v8f;

__device__ __forceinline__ void async_b128_to_lds(unsigned lds_addr, unsigned voff,
                                                  unsigned long long sbase) {
  // GVS mode: mem = SGPR64 + VGPR32 offset; VDST = LDS byte address.
  asm volatile("global_load_async_to_lds_b128 %0, %1, %2"
               :: "v"(lds_addr), "v"(voff), "s"(sbase)
               : "memory");
}

__global__ __launch_bounds__(256) void vae_metrics_kernel(
    const float* __restrict__ pred, const float* __restrict__ truth,
    const int* __restrict__ mask, const float* __restrict__ obsrv_std,
    float* __restrict__ out_logden, float* __restrict__ ws_mse) {
  __shared__ float smem[NSTAGE * STAGE_FLOATS];  // 20480 floats = 80 KB

  const int tid = threadIdx.x;
  const int b = blockIdx.x;

  const unsigned long long truth_b =
      (unsigned long long)(const void*)(truth + (size_t)b * TD_);
  const unsigned long long mask_b =
      (unsigned long long)(const void*)(mask + (size_t)b * TD_);
  const unsigned long long pb0 =
      (unsigned long long)(const void*)(pred + ((size_t)0 * B_ + b) * TD_);
  const unsigned long long pb1 =
      (unsigned long long)(const void*)(pred + ((size_t)1 * B_ + b) * TD_);
  const unsigned long long pb2 =
      (unsigned long long)(const void*)(pred + ((size_t)2 * B_ + b) * TD_);

  const unsigned lds0 = (unsigned)(uintptr_t)(&smem[0]);

  // Thread mapping within a tile: trow = tid/16, d = 4*(tid%16).
  // Byte offset of this thread's float4 inside a tile slab == tid*16.
  auto issue_tile = [&](int tile) {
    const int stage = tile % NSTAGE;
    const unsigned l = lds0 + (unsigned)(stage * STAGE_FLOATS * 4) + (unsigned)(tid * 16);
    const unsigned voff = (unsigned)(tile * 4096 + tid * 16);
    async_b128_to_lds(l,             voff, truth_b);
    async_b128_to_lds(l + 4096u,     voff, mask_b);
    async_b128_to_lds(l + 8192u,     voff, pb0);
    async_b128_to_lds(l + 12288u,    voff, pb1);
    async_b128_to_lds(l + 16384u,    voff, pb2);
  };

  float acc[3][4] = {{0.f, 0.f, 0.f, 0.f}, {0.f, 0.f, 0.f, 0.f}, {0.f, 0.f, 0.f, 0.f}};
  float cnt[4] = {0.f, 0.f, 0.f, 0.f};

  issue_tile(0);
  issue_tile(1);
  issue_tile(2);

  for (int t = 0; t < NTILE; ++t) {
    if (t + 3 < NTILE) issue_tile(t + 3);
    // 5 async loads per tile per thread; ensure tile t's loads are complete.
    if (t <= NTILE - 4) {
      asm volatile("s_wait_asynccnt 15" ::: "memory");
    } else if (t == NTILE - 3) {
      asm volatile("s_wait_asynccnt 10" ::: "memory");
    } else if (t == NTILE - 2) {
      asm volatile("s_wait_asynccnt 5" ::: "memory");
    } else {
      asm volatile("s_wait_asynccnt 0" ::: "memory");
    }
    __syncthreads();

    const float* sb = &smem[(t % NSTAGE) * STAGE_FLOATS];
    const float4 tr = *(const float4*)(sb + tid * 4);
    const int4 mi = *(const int4*)(sb + 1024 + tid * 4);
    const float mf0 = (float)mi.x, mf1 = (float)mi.y, mf2 = (float)mi.z, mf3 = (float)mi.w;
    cnt[0] += mf0; cnt[1] += mf1; cnt[2] += mf2; cnt[3] += mf3;

#pragma unroll
    for (int s = 0; s < 3; ++s) {
      const float4 p = *(const float4*)(sb + 2048 + s * 1024 + tid * 4);
      float dx = p.x - tr.x;
      float dy = p.y - tr.y;
      float dz = p.z - tr.z;
      float dw = p.w - tr.w;
      acc[s][0] = fmaf(mf0 * dx, dx, acc[s][0]);
      acc[s][1] = fmaf(mf1 * dy, dy, acc[s][1]);
      acc[s][2] = fmaf(mf2 * dz, dz, acc[s][2]);
      acc[s][3] = fmaf(mf3 * dw, dw, acc[s][3]);
    }
    __syncthreads();
  }

  // Cross-trow reduction staging: red[col][trow], col = type*64 + d,
  // types 0..2 = sq_sum per s, type 3 = cnt. 256 cols x 16 trows = 16 KB,
  // aliased over the (now dead) stage buffers.
  float* red = smem;
  {
    const int trow = tid >> 4;
    const int dbase = (tid & 15) << 2;
#pragma unroll
    for (int c = 0; c < 4; ++c) {
      red[(0 * 64 + dbase + c) * 16 + trow] = acc[0][c];
      red[(1 * 64 + dbase + c) * 16 + trow] = acc[1][c];
      red[(2 * 64 + dbase + c) * 16 + trow] = acc[2][c];
      red[(192 + dbase + c) * 16 + trow] = cnt[c];
    }
  }
  __syncthreads();

  // Column-sum via WMMA: ones(16x4) x B(4x16) accumulated over 4 K-chunks
  // reduces 16 trow partials for 16 columns per group. With A == ones the
  // result is independent of which K-slot each trow occupies, so only the
  // N==lane mapping matters. Wave 0 handles all 16 column groups.
  float* csum = smem + 4096;  // 256 floats
  if (tid < 32) {
    const int n = tid & 15;
    const int hi = tid >> 4;
    v2f a;
    a[0] = 1.0f;
    a[1] = 1.0f;
    for (int g = 0; g < 16; ++g) {
      v8f c = {};
#pragma unroll
      for (int k = 0; k < 4; ++k) {
        v2f bb;
        bb[0] = red[(g * 16 + n) * 16 + (4 * k + hi)];
        bb[1] = red[(g * 16 + n) * 16 + (4 * k + 2 + hi)];
        c = __builtin_amdgcn_wmma_f32_16x16x4_f32(
            /*neg_a=*/false, a, /*neg_b=*/false, bb,
            /*c_mod=*/(short)0, c, /*reuse_a=*/false, /*reuse_b=*/false);
      }
      if (tid < 16) csum[g * 16 + tid] = c[0];  // row M=0: lane n holds col g*16+n
    }
  }
  __syncthreads();

  // Epilogue: per-d log-prob and mse terms, then block reduction.
  float* lpbuf = smem + 4352;  // 64 floats
  float* msbuf = smem + 4416;  // 64 floats
  if (tid < 64) {
    const float c0 = csum[192 + tid];
    float lp = 0.f, ms = 0.f;
    if (c0 > 0.f) {
      const float stdv = obsrv_std[0];
      const float inv = 1.0f / c0;
      const float k1 = -0.5f / (stdv * stdv);
      const float kc = -logf(stdv) - 0.5f * 1.8378770664093453f;  // -log(std) - 0.5*log(2pi)
#pragma unroll
      for (int s = 0; s < 3; ++s) {
        const float ss = csum[s * 64 + tid];
        ms += ss * inv;             // mse_pd summed over s
        lp += k1 * ss * inv + kc;   // logpdf_sum / cnt
      }
    }
    lpbuf[tid] = lp;
    msbuf[tid] = ms;
  }
  __syncthreads();
  if (tid == 0) {
    float lps = 0.f, mss = 0.f;
    for (int d = 0; d < 64; ++d) {
      lps += lpbuf[d];
      mss += msbuf[d];
    }
    out_logden[b] = lps * (1.0f / (3.0f * 64.0f));  // mean over s and d
    ws_mse[b] = mss;                                 // partial sum over s,d
  }
}

__global__ __launch_bounds__(256) void vae_mse_finalize(const float* __restrict__ ws,
                                                        float* __restrict__ out) {
  __shared__ float r[256];
  r[threadIdx.x] = ws[threadIdx.x];
  __syncthreads();
  for (int off = 128; off > 0; off >>= 1) {
    if (threadIdx.x < (unsigned)off) r[threadIdx.x] += r[threadIdx.x + off];
    __syncthreads();
  }
  if (threadIdx.x == 0) out[B_] = r[0] * (1.0f / 49152.0f);  // / (S*B*D)
}

extern "C" void kernel_launch(void* const* d_in, const int* in_sizes, int n_in,
                              void* d_out, int out_size, void* d_ws, size_t ws_size,
                              hipStream_t stream) {
  const float* pred = (const float*)d_in[0];
  const float* truth = (const float*)d_in[1];
  const int* mask = (const int*)d_in[2];
  const float* stdp = (const float*)d_in[3];
  float* out = (float*)d_out;   // [0..255] log_density, [256] mse
  float* ws = (float*)d_ws;     // 256 floats of mse partials

  vae_metrics_kernel<<<B_, 256, 0, stream>>>(pred, truth, mask, stdp, out, ws);
  vae_mse_finalize<<<1, 256, 0, stream>>>(ws, out);
}